// GraphConvolution_59081570123777
// MI455X (gfx1250) — compile-verified
//
#include <hip/hip_runtime.h>

#define N_NODES 100000
#define N_EDGES 640000
#define FDIM    128

typedef __attribute__((ext_vector_type(2))) float v2f;
typedef __attribute__((ext_vector_type(8))) float v8f;

// A LDS stride (floats): multiple of 4 (16B-aligned float4 stores, 8B-aligned
// float2 fragment reads) and 36*m mod 64 distinct for m=0..15 -> conflict-free
// ds_load_b64 A-fragment fetch.
#define A_STRIDE 36
// B LDS pair-row stride (floats): B stored K-pair-swizzled,
//   Bs[(k>>1)*B_STRIDE + col*2 + (k&1)]
// so one ds_load_b64 yields a full B fragment (K, K+1 of one column).
// 288 mod 64 == 32 -> the two half-wave pair-rows hit disjoint bank ranges.
#define B_STRIDE 288

// ---------------------------------------------------------------------------
// support[N,128] = input[N,128] @ weight[128,128], exact fp32 via
// V_WMMA_F32_16X16X4_F32. Block = 256 threads (8 waves) -> 128x128 panel,
// 16 rows x 128 cols per wave (8 accumulator tiles, 64 VGPRs).
// ---------------------------------------------------------------------------
__global__ __launch_bounds__(256) void gemm_wmma_f32(const float* __restrict__ in,
                                                     const float* __restrict__ wt,
                                                     float* __restrict__ sup) {
  __shared__ float As[128 * A_STRIDE];   // 18.0 KB
  __shared__ float Bs[16 * B_STRIDE];    // 18.0 KB (16 K-pairs x 128 cols x 2)
  const int t    = threadIdx.x;
  const int wave = t >> 5;
  const int lane = t & 31;
  const int half = lane >> 4;   // 0: lanes 0-15, 1: lanes 16-31
  const int m    = lane & 15;
  const int rowBase = blockIdx.x * 128;

  v8f acc[8] = {};

  for (int k0 = 0; k0 < FDIM; k0 += 32) {
    __syncthreads();
    // Stage A chunk: 128 rows x 32 K (coalesced float4 global loads)
    #pragma unroll
    for (int p = 0; p < 4; ++p) {
      const int r  = p * 32 + (t >> 3);   // 0..127
      const int j  = (t & 7) * 4;         // 0..28
      const int gr = rowBase + r;
      float4 v = make_float4(0.f, 0.f, 0.f, 0.f);
      if (gr < N_NODES) v = *(const float4*)(in + (long long)gr * FDIM + k0 + j);
      *(float4*)(&As[r * A_STRIDE + j]) = v;
    }
    // Stage B chunk (32 K x 128 cols), K-pair swizzled for b64 fragment reads.
    #pragma unroll
    for (int p = 0; p < 4; ++p) {
      const int kr = p * 8 + (t >> 5);    // 0..31
      const int j  = (t & 31) * 4;        // 0..124
      float4 v = *(const float4*)(wt + (k0 + kr) * FDIM + j);
      float* bb = &Bs[(kr >> 1) * B_STRIDE + j * 2 + (kr & 1)];
      bb[0] = v.x; bb[2] = v.y; bb[4] = v.z; bb[6] = v.w;
    }
    __syncthreads();

    #pragma unroll
    for (int ks = 0; ks < 8; ++ks) {
      const int kk = ks * 4;
      // A fragment (ISA 16x4 f32 layout): lanes 0-15 hold K=kk,kk+1 of row m;
      // lanes 16-31 hold K=kk+2,kk+3. Single 8B-aligned ds_load_b64.
      v2f a = *(const v2f*)(&As[(wave * 16 + m) * A_STRIDE + kk + 2 * half]);
      const float* brow = &Bs[(ks * 2 + half) * B_STRIDE];
      #pragma unroll
      for (int n = 0; n < 8; ++n) {
        // B fragment: K=kk+2*half (.x) and K=kk+2*half+1 (.y) of column
        // n*16+m -- adjacent in the swizzled layout: one ds_load_b64.
        v2f b = *(const v2f*)(brow + (n * 16 + m) * 2);
        acc[n] = __builtin_amdgcn_wmma_f32_16x16x4_f32(
            false, a, false, b, (short)0, acc[n], false, false);
      }
    }
  }

  // C/D layout: VGPR i = row (i + 8*half) of the 16x16 tile, col = m.
  const int waveRow = rowBase + wave * 16;
  #pragma unroll
  for (int n = 0; n < 8; ++n) {
    #pragma unroll
    for (int i = 0; i < 8; ++i) {
      const int r = waveRow + half * 8 + i;
      if (r < N_NODES) sup[(long long)r * FDIM + n * 16 + m] = acc[n][i];
    }
  }
}

// ---------------------------------------------------------------------------
// out[rows[e]] += support[cols[e]]  — one wave32 per edge, float4 per lane,
// native fp32 L2 atomics. support/out are L2-resident (51 MB each, 192 MB L2).
// ---------------------------------------------------------------------------
__global__ __launch_bounds__(256) void spmm_scatter(const float* __restrict__ sup,
                                                    const long long* __restrict__ adj,
                                                    float* __restrict__ out) {
  const long long e = (long long)blockIdx.x * 8 + (threadIdx.x >> 5);
  if (e >= N_EDGES) return;
  const int lane = threadIdx.x & 31;
  const int row = (int)adj[e];             // adj[0][e]
  const int col = (int)adj[N_EDGES + e];   // adj[1][e]
  float4 v = *(const float4*)(sup + (long long)col * FDIM + lane * 4);
  float* dst = out + (long long)row * FDIM + lane * 4;
  unsafeAtomicAdd(dst + 0, v.x);
  unsafeAtomicAdd(dst + 1, v.y);
  unsafeAtomicAdd(dst + 2, v.z);
  unsafeAtomicAdd(dst + 3, v.w);
}

// ---------------------------------------------------------------------------
// In-place ReLU, float4 vectorized. (out is re-zeroed at the start of every
// kernel_launch, so in-place RMW is replay-safe.)
// ---------------------------------------------------------------------------
__global__ __launch_bounds__(256) void relu_kernel(float* __restrict__ out) {
  const long long i = ((long long)blockIdx.x * blockDim.x + threadIdx.x) * 4;
  if (i < (long long)N_NODES * FDIM) {
    float4 v = *(float4*)(out + i);
    v.x = fmaxf(v.x, 0.f);
    v.y = fmaxf(v.y, 0.f);
    v.z = fmaxf(v.z, 0.f);
    v.w = fmaxf(v.w, 0.f);
    *(float4*)(out + i) = v;
  }
}

extern "C" void kernel_launch(void* const* d_in, const int* in_sizes, int n_in,
                              void* d_out, int out_size, void* d_ws, size_t ws_size,
                              hipStream_t stream) {
  const float* in       = (const float*)d_in[0];       // [N,128] fp32
  const long long* adj  = (const long long*)d_in[1];   // [2,E] int64
  const float* wt       = (const float*)d_in[2];       // [128,128] fp32
  float* out            = (float*)d_out;               // [N,128] fp32
  float* sup            = (float*)d_ws;                // workspace: [N,128] fp32

  hipMemsetAsync(d_out, 0, (size_t)N_NODES * FDIM * sizeof(float), stream);

  gemm_wmma_f32<<<(N_NODES + 127) / 128, 256, 0, stream>>>(in, wt, sup);
  spmm_scatter<<<(N_EDGES + 7) / 8, 256, 0, stream>>>(sup, adj, out);

  const long long nq = (long long)N_NODES * FDIM / 4;
  relu_kernel<<<(unsigned)((nq + 255) / 256), 256, 0, stream>>>(out);
}